// _YOLO_50199577756153
// MI455X (gfx1250) — compile-verified
//
#include <hip/hip_runtime.h>

// ---------------------------------------------------------------------------
// YOLO loss for MI455X (gfx1250). Compile-only target.
//   out[0] = 1.0 * obj_loss   / B
//   out[1] = 0.5 * noobj_loss / B
//   out[2] = 5.0 * bbox_loss  / B
//   out[3] = 1.0 * clf_loss   / B
// ---------------------------------------------------------------------------

#define LAMBDA_OBJ    1.0f
#define LAMBDA_NOOBJ  0.5f
#define LAMBDA_CLASS  1.0f
#define LAMBDA_COORDS 5.0f

constexpr int Bn = 64, Hn = 52, Wn = 52, An = 5, Cn = 80, Mn = 64;
constexpr int NCELL = Bn * Hn * Wn * An;   // 865280 anchor slots
constexpr int NOOBJ_BLOCKS = 512;

typedef float v2f __attribute__((ext_vector_type(2)));
typedef float v8f __attribute__((ext_vector_type(8)));

// Deterministic fixed-order tree sum over 64 shared floats; result in a[0].
// Must be called by ALL threads of the (128-thread) block.
__device__ __forceinline__ void sum64(float* a) {
    const int t = threadIdx.x;
    __syncthreads();
    if (t < 32) a[t] += a[t + 32];
    __syncthreads();
    if (t < 16) a[t] += a[t + 16];
    __syncthreads();
    if (t < 8)  a[t] += a[t + 8];
    __syncthreads();
    if (t < 4)  a[t] += a[t + 4];
    __syncthreads();
    if (t < 2)  a[t] += a[t + 2];
    __syncthreads();
    if (t < 1)  a[t] += a[t + 1];
    __syncthreads();
}

// One block per batch image b. 128 threads = 4 wave32.
// Threads 0..63: per-GT-box match work (IoU, argmax, obj/xy/wh terms, mask).
// All 4 waves:   WMMA row-sum-of-squares over the 400 gathered scores/box.
__global__ __launch_bounds__(128)
void yolo_match_kernel(const float* __restrict__ pred_boxes,
                       const float* __restrict__ pred_o,
                       const float* __restrict__ pred_scores,
                       const float* __restrict__ gt_boxes,
                       const int*   __restrict__ gt_labels,
                       unsigned char* __restrict__ mask,
                       float* __restrict__ obj_part,
                       float* __restrict__ clf_part,
                       float* __restrict__ bbox_part) {
    __shared__ int   sbase[Mn];     // float-index of each GT box's 400-score row
    __shared__ float s_cross[Mn];   // sum_a s[a, label]
    __shared__ float s_row2[Mn];    // WMMA: sum_{a,c} s^2
    __shared__ float s_obj[Mn], s_xy[Mn], s_clf[Mn];
    __shared__ float s_ssw[Mn], s_ssh[Mn], s_ss2w[Mn], s_ss2h[Mn];
    __shared__ float s_sgw[Mn], s_sgh[Mn], s_sg2w[Mn], s_sg2h[Mn];

    const int b = blockIdx.x;
    const int t = threadIdx.x;

    if (t < Mn) {
        const int m = t;
        const float g0 = gt_boxes[((b * Mn) + m) * 4 + 0];
        const float g1 = gt_boxes[((b * Mn) + m) * 4 + 1];
        const float g2 = gt_boxes[((b * Mn) + m) * 4 + 2];
        const float g3 = gt_boxes[((b * Mn) + m) * 4 + 3];
        const float gcx = (g0 + g2) * 0.5f, gcy = (g1 + g3) * 0.5f;
        const float gw = g2 - g0,          gh = g3 - g1;
        const int gx = (int)floorf(gcx * (float)Wn);
        const int gy = (int)floorf(gcy * (float)Hn);
        const int cellbase = ((b * Hn + gy) * Wn + gx) * An;

        float best_iou = -1.0f;
        int   best_a = 0;
        float selx = 0.f, sely = 0.f, selw = 0.f, selh = 0.f;
#pragma unroll
        for (int a = 0; a < An; ++a) {
            const float* pb = pred_boxes + (size_t)(cellbase + a) * 4;
            const float px = (pb[0] + (float)gx) / (float)Wn;
            const float py = (pb[1] + (float)gy) / (float)Hn;
            const float hw = pb[2] * 0.5f, hh = pb[3] * 0.5f;
            const float x1 = px - hw, y1 = py - hh;
            const float x2 = px + hw, y2 = py + hh;
            const float ltx = fmaxf(g0, x1), lty = fmaxf(g1, y1);
            const float rbx = fminf(g2, x2), rby = fminf(g3, y2);
            const float iw = fmaxf(rbx - ltx, 0.f), ih = fmaxf(rby - lty, 0.f);
            const float inter = iw * ih;
            const float a1 = gw * gh;
            const float a2 = (x2 - x1) * (y2 - y1);
            const float iou = inter / (a1 + a2 - inter);
            if (iou > best_iou) {   // strict '>' == argmax first-max tie-break
                best_iou = iou; best_a = a;
                selx = (x1 + x2) * 0.5f; sely = (y1 + y2) * 0.5f;
                selw = pb[2];            selh = pb[3];
            }
        }
        const float osel = pred_o[cellbase + best_a];
        const float od = best_iou - osel;
        s_obj[m] = od * od;
        mask[cellbase + best_a] = 1;     // duplicates write the same value: dedup-safe

        const float dx = gcx - selx, dy = gcy - sely;
        s_xy[m] = dx * dx + dy * dy;

        const float sqw = sqrtf(selw), sqh = sqrtf(selh);
        const float sgw = sqrtf(gw),   sgh = sqrtf(gh);
        s_ssw[m] = sqw;  s_ssh[m] = sqh;  s_ss2w[m] = sqw * sqw;  s_ss2h[m] = sqh * sqh;
        s_sgw[m] = sgw;  s_sgh[m] = sgh;  s_sg2w[m] = sgw * sgw;  s_sg2h[m] = sgh * sgh;

        const int rowbase = cellbase * Cn;   // start of 400 contiguous scores
        sbase[m] = rowbase;
        const int label = gt_labels[b * Mn + m];
        float cr = 0.f;
#pragma unroll
        for (int a = 0; a < An; ++a) cr += pred_scores[rowbase + a * Cn + label];
        s_cross[m] = cr;
    }
    __syncthreads();

    // --- WMMA phase: D += (A .* A) x ones, K accumulated in steps of 4. ---
    // A (16x4 f32): lanes 0-15 carry rows, VGPR0/1 = K0/K1; lanes 16-31 = K2/K3.
    // B = ones(4x16)  =>  D[m,n] = row-sum for all n.
    {
        const int wave = t >> 5;                 // 0..3, 16 GT rows each
        const int lane = t & 31;
        const int row  = wave * 16 + (lane & 15);
        const int koff = (lane < 16) ? 0 : 2;
        const float* rowp = pred_scores + sbase[row] + koff;

        v8f acc = {0.f, 0.f, 0.f, 0.f, 0.f, 0.f, 0.f, 0.f};
        v2f ones; ones.x = 1.0f; ones.y = 1.0f;
#pragma unroll 4
        for (int k = 0; k < An * Cn; k += 4) {   // 100 WMMA steps
            v2f a = *(const v2f*)(rowp + k);     // 8B-aligned (rowbase % 80 == 0)
            a.x *= a.x;
            a.y *= a.y;
            acc = __builtin_amdgcn_wmma_f32_16x16x4_f32(
                /*neg_a=*/false, a, /*neg_b=*/false, ones,
                /*c_mod=*/(short)0, acc, /*reuse_a=*/false, /*reuse_b=*/false);
        }
        // D layout: lane0 v[r] = rowsum(wave*16+r), lane16 v[r] = rowsum(wave*16+8+r)
        if (lane == 0) {
#pragma unroll
            for (int r = 0; r < 8; ++r) s_row2[wave * 16 + r] = acc[r];
        }
        if (lane == 16) {
#pragma unroll
            for (int r = 0; r < 8; ++r) s_row2[wave * 16 + 8 + r] = acc[r];
        }
    }
    __syncthreads();

    // clf per box: sum s^2 - 2*sum_a s[a,label] + A   (onehot expansion)
    if (t < Mn) s_clf[t] = s_row2[t] - 2.0f * s_cross[t] + (float)An;

    sum64(s_obj);  sum64(s_xy);   sum64(s_clf);
    sum64(s_ssw);  sum64(s_ssh);  sum64(s_ss2w); sum64(s_ss2h);
    sum64(s_sgw);  sum64(s_sgh);  sum64(s_sg2w); sum64(s_sg2h);

    if (t == 0) {
        // MxM cross term: sum_{i,j,k}(g_jk - s_ik)^2
        //   = sum_k [ M*Sum(g^2) + M*Sum(s^2) - 2*Sum(g)*Sum(s) ]
        const float wh = (float)Mn * (s_sg2w[0] + s_sg2h[0])
                       + (float)Mn * (s_ss2w[0] + s_ss2h[0])
                       - 2.0f * (s_sgw[0] * s_ssw[0] + s_sgh[0] * s_ssh[0]);
        obj_part[b]  = s_obj[0];
        clf_part[b]  = s_clf[0];
        bbox_part[b] = s_xy[0] + wh;
    }
}

// Streaming pass: noobj = sum pred_o^2 where mask == 0. pred_o/mask are dead
// after this -> non-temporal loads to avoid polluting L2 with 5 MB of trash.
__global__ __launch_bounds__(256)
void yolo_noobj_kernel(const float* __restrict__ pred_o,
                       const unsigned char* __restrict__ mask,
                       float* __restrict__ part, int n) {
    __shared__ float sm[256];
    float acc = 0.f;
    const int stride = gridDim.x * blockDim.x;
    for (int i = blockIdx.x * blockDim.x + threadIdx.x; i < n; i += stride) {
        const float o = __builtin_nontemporal_load(pred_o + i);
        const unsigned char mk = __builtin_nontemporal_load(mask + i);
        acc += mk ? 0.f : o * o;
    }
    sm[threadIdx.x] = acc;
    __syncthreads();
#pragma unroll
    for (int s = 128; s > 0; s >>= 1) {
        if ((int)threadIdx.x < s) sm[threadIdx.x] += sm[threadIdx.x + s];
        __syncthreads();
    }
    if (threadIdx.x == 0) part[blockIdx.x] = sm[0];
}

// Final combine: tiny, strictly serial per output -> bitwise deterministic.
__global__ void yolo_combine_kernel(const float* __restrict__ obj,
                                    const float* __restrict__ clf,
                                    const float* __restrict__ bbox,
                                    const float* __restrict__ noobj,
                                    float* __restrict__ out) {
    const int t = threadIdx.x;
    if (t == 0) {
        float s = 0.f;
        for (int i = 0; i < Bn; ++i) s += obj[i];
        out[0] = LAMBDA_OBJ * s / (float)Bn;
    } else if (t == 1) {
        float s = 0.f;
        for (int i = 0; i < NOOBJ_BLOCKS; ++i) s += noobj[i];
        out[1] = LAMBDA_NOOBJ * s / (float)Bn;
    } else if (t == 2) {
        float s = 0.f;
        for (int i = 0; i < Bn; ++i) s += bbox[i];
        out[2] = LAMBDA_COORDS * s / (float)Bn;
    } else if (t == 3) {
        float s = 0.f;
        for (int i = 0; i < Bn; ++i) s += clf[i];
        out[3] = LAMBDA_CLASS * s / (float)Bn;
    }
}

extern "C" void kernel_launch(void* const* d_in, const int* in_sizes, int n_in,
                              void* d_out, int out_size, void* d_ws, size_t ws_size,
                              hipStream_t stream) {
    const float* pred_boxes  = (const float*)d_in[0];
    const float* pred_o      = (const float*)d_in[1];
    const float* pred_scores = (const float*)d_in[2];
    const float* gt_boxes    = (const float*)d_in[3];
    const int*   gt_labels   = (const int*)d_in[4];
    float* out = (float*)d_out;

    // Workspace layout (NCELL is a multiple of 256 -> float region aligned).
    unsigned char* mask     = (unsigned char*)d_ws;
    float* obj_part   = (float*)((char*)d_ws + NCELL);
    float* clf_part   = obj_part + Bn;
    float* bbox_part  = clf_part + Bn;
    float* noobj_part = bbox_part + Bn;   // NOOBJ_BLOCKS floats

    hipMemsetAsync(mask, 0, NCELL, stream);   // capturable memset node

    yolo_match_kernel<<<Bn, 128, 0, stream>>>(
        pred_boxes, pred_o, pred_scores, gt_boxes, gt_labels,
        mask, obj_part, clf_part, bbox_part);

    yolo_noobj_kernel<<<NOOBJ_BLOCKS, 256, 0, stream>>>(
        pred_o, mask, noobj_part, NCELL);

    yolo_combine_kernel<<<1, 32, 0, stream>>>(
        obj_part, clf_part, bbox_part, noobj_part, out);
}